// TripletLoss_87033217286636
// MI455X (gfx1250) — compile-verified
//
#include <hip/hip_runtime.h>
#include <hip/hip_bf16.h>
#include <math.h>

#define NROWS 8192
#define DDIM 128
#define NTILES (NROWS / 16)      // 512 tiles of 16 rows
#define S_SLICES 8               // j-range split per i-tile
#define JT_PER_SLICE (NTILES / S_SLICES)  // 64
#define MARGIN 0.5f

typedef float v2f __attribute__((ext_vector_type(2)));
typedef float v8f __attribute__((ext_vector_type(8)));

// ---------------- Kernel 1: row squared norms ----------------
__global__ void sq_kernel(const float* __restrict__ E, float* __restrict__ sq) {
    int r = blockIdx.x * blockDim.x + threadIdx.x;
    if (r >= NROWS) return;
    const float* e = E + (size_t)r * DDIM;
    float s = 0.0f;
    #pragma unroll 4
    for (int k = 0; k < DDIM; ++k) s += e[k] * e[k];
    sq[r] = s;
}

// ---------------- Kernel 2: fused WMMA Gram + batch-hard mining ----------------
// One wave (32 lanes) per (i-tile, j-slice). C tile layout (f32 16x16):
//   lane L: column N = L&15 ; VGPR m holds row M = m + 8*(L>>4).
// A tile layout (f32 16x4): lane L: row M = L&15 ; half (L>>4) selects K pair.
// B (4x16) mirrors A with N = L&15. Gram symmetry => identical addressing for A/B.
__global__ __launch_bounds__(32)
void mine_kernel(const float* __restrict__ E, const int* __restrict__ labels,
                 const float* __restrict__ sq,
                 float* __restrict__ pPosV, int* __restrict__ pPosI,
                 float* __restrict__ pNegV, int* __restrict__ pNegI) {
    const int it    = blockIdx.x;   // 0..511
    const int slice = blockIdx.y;   // 0..7
    const int lane  = threadIdx.x;  // 0..31 (wave32)
    const int half  = lane >> 4;
    const int l15   = lane & 15;

    // Preload full A fragments for this i-tile (16x128): 64 floats per lane.
    const int arow = it * 16 + l15;
    float aA[64];
    const float* Ar = E + (size_t)arow * DDIM + half * 2;
    #pragma unroll
    for (int k = 0; k < 32; ++k) {
        aA[2 * k]     = Ar[4 * k];
        aA[2 * k + 1] = Ar[4 * k + 1];
    }

    // Rows this lane owns in the C layout.
    const int rbase = it * 16 + half * 8;
    int rowLab[8]; float rowSq[8];
    #pragma unroll
    for (int m = 0; m < 8; ++m) { rowLab[m] = labels[rbase + m]; rowSq[m] = sq[rbase + m]; }

    float posV[8], negV[8]; int posI[8], negI[8];
    #pragma unroll
    for (int m = 0; m < 8; ++m) {
        posV[m] = -INFINITY; posI[m] = 0;
        negV[m] =  INFINITY; negI[m] = 0;
    }

    const int jt0 = slice * JT_PER_SLICE;
    for (int jt = jt0; jt < jt0 + JT_PER_SLICE; ++jt) {
        const int col    = jt * 16 + l15;
        const int colLab = labels[col];
        const float colSq = sq[col];

        const float* Br = E + (size_t)col * DDIM + half * 2;
        // 4 independent accumulators to pipeline the WMMA RAW chain.
        v8f c0 = {}, c1 = {}, c2 = {}, c3 = {};
        #pragma unroll
        for (int k = 0; k < 32; k += 4) {
            v2f a0, a1, a2, a3, b0, b1, b2, b3;
            a0.x = aA[2*k+0]; a0.y = aA[2*k+1];
            a1.x = aA[2*k+2]; a1.y = aA[2*k+3];
            a2.x = aA[2*k+4]; a2.y = aA[2*k+5];
            a3.x = aA[2*k+6]; a3.y = aA[2*k+7];
            b0.x = Br[4*k+0];  b0.y = Br[4*k+1];
            b1.x = Br[4*k+4];  b1.y = Br[4*k+5];
            b2.x = Br[4*k+8];  b2.y = Br[4*k+9];
            b3.x = Br[4*k+12]; b3.y = Br[4*k+13];
            c0 = __builtin_amdgcn_wmma_f32_16x16x4_f32(false, a0, false, b0, (short)0, c0, false, false);
            c1 = __builtin_amdgcn_wmma_f32_16x16x4_f32(false, a1, false, b1, (short)0, c1, false, false);
            c2 = __builtin_amdgcn_wmma_f32_16x16x4_f32(false, a2, false, b2, (short)0, c2, false, false);
            c3 = __builtin_amdgcn_wmma_f32_16x16x4_f32(false, a3, false, b3, (short)0, c3, false, false);
        }

        #pragma unroll
        for (int m = 0; m < 8; ++m) {
            const float g  = (c0[m] + c1[m]) + (c2[m] + c3[m]);
            const float d2 = rowSq[m] + colSq - 2.0f * g;
            const int r = rbase + m;
            if (rowLab[m] == colLab) {
                if ((r != col) && (d2 > posV[m])) { posV[m] = d2; posI[m] = col; }
            } else {
                if (d2 < negV[m]) { negV[m] = d2; negI[m] = col; }
            }
        }
    }

    // Cross-lane argmax/argmin within each 16-lane group (masks keep bit4 fixed).
    #pragma unroll
    for (int m = 0; m < 8; ++m) {
        #pragma unroll
        for (int d = 1; d < 16; d <<= 1) {
            float ov = __shfl_xor(posV[m], d, 32);
            int   oi = __shfl_xor(posI[m], d, 32);
            if (ov > posV[m] || (ov == posV[m] && oi < posI[m])) { posV[m] = ov; posI[m] = oi; }
            float nv = __shfl_xor(negV[m], d, 32);
            int   ni = __shfl_xor(negI[m], d, 32);
            if (nv < negV[m] || (nv == negV[m] && ni < negI[m])) { negV[m] = nv; negI[m] = ni; }
        }
    }

    if (l15 == 0) {
        #pragma unroll
        for (int m = 0; m < 8; ++m) {
            const int o = (rbase + m) * S_SLICES + slice;
            pPosV[o] = posV[m]; pPosI[o] = posI[m];
            pNegV[o] = negV[m]; pNegI[o] = negI[m];
        }
    }
}

// ---------------- Kernel 3: combine slice partials ----------------
__global__ void combine_kernel(const float* __restrict__ pPosV, const int* __restrict__ pPosI,
                               const float* __restrict__ pNegV, const int* __restrict__ pNegI,
                               int* __restrict__ posIdx, int* __restrict__ negIdx) {
    int r = blockIdx.x * blockDim.x + threadIdx.x;
    if (r >= NROWS) return;
    float pv = -INFINITY; int pi = 0;
    float nv =  INFINITY; int ni = 0;
    #pragma unroll
    for (int s = 0; s < S_SLICES; ++s) {
        float v = pPosV[r * S_SLICES + s]; int i = pPosI[r * S_SLICES + s];
        if (v > pv || (v == pv && i < pi)) { pv = v; pi = i; }
        v = pNegV[r * S_SLICES + s]; i = pNegI[r * S_SLICES + s];
        if (v < nv || (v == nv && i < ni)) { nv = v; ni = i; }
    }
    posIdx[r] = pi; negIdx[r] = ni;
}

// ---------------- Kernel 4: exact ap/an recompute + per-row hinge ----------------
__global__ void loss_kernel(const float* __restrict__ E, const int* __restrict__ posIdx,
                            const int* __restrict__ negIdx, float* __restrict__ rowLoss) {
    int r = blockIdx.x * blockDim.x + threadIdx.x;
    if (r >= NROWS) return;
    const float* ei = E + (size_t)r * DDIM;
    const float* ep = E + (size_t)posIdx[r] * DDIM;
    const float* en = E + (size_t)negIdx[r] * DDIM;
    float ap = 0.0f, an = 0.0f;
    #pragma unroll 4
    for (int k = 0; k < DDIM; ++k) {
        float dp = ei[k] - ep[k]; ap += dp * dp;
        float dn = ei[k] - en[k]; an += dn * dn;
    }
    float l = ap - an + MARGIN;
    rowLoss[r] = l > 0.0f ? l : 0.0f;
}

// ---------------- Kernel 5: deterministic mean reduction ----------------
__global__ void reduce_kernel(const float* __restrict__ rowLoss, float* __restrict__ out) {
    __shared__ float sm[256];
    int t = threadIdx.x;
    float s = 0.0f;
    for (int i = t; i < NROWS; i += 256) s += rowLoss[i];
    sm[t] = s;
    __syncthreads();
    for (int w = 128; w > 0; w >>= 1) {
        if (t < w) sm[t] += sm[t + w];
        __syncthreads();
    }
    if (t == 0) out[0] = sm[0] * (1.0f / (float)NROWS);
}

extern "C" void kernel_launch(void* const* d_in, const int* in_sizes, int n_in,
                              void* d_out, int out_size, void* d_ws, size_t ws_size,
                              hipStream_t stream) {
    const float* E      = (const float*)d_in[0];   // [8192,128] f32
    const int*   labels = (const int*)d_in[1];     // [8192] i32

    char* ws = (char*)d_ws;
    float* sq      = (float*)(ws);                               // 32 KB
    float* pPosV   = (float*)(ws + 32768);                       // 256 KB
    int*   pPosI   = (int*)  (ws + 32768 + 262144);              // 256 KB
    float* pNegV   = (float*)(ws + 32768 + 2 * 262144);          // 256 KB
    int*   pNegI   = (int*)  (ws + 32768 + 3 * 262144);          // 256 KB
    int*   posIdx  = (int*)  (ws + 32768 + 4 * 262144);          // 32 KB
    int*   negIdx  = (int*)  (ws + 32768 + 4 * 262144 + 32768);  // 32 KB
    float* rowLoss = (float*)(ws + 32768 + 4 * 262144 + 65536);  // 32 KB

    sq_kernel<<<NROWS / 256, 256, 0, stream>>>(E, sq);

    dim3 grid(NTILES, S_SLICES);
    mine_kernel<<<grid, 32, 0, stream>>>(E, labels, sq, pPosV, pPosI, pNegV, pNegI);

    combine_kernel<<<NROWS / 256, 256, 0, stream>>>(pPosV, pPosI, pNegV, pNegI, posIdx, negIdx);
    loss_kernel<<<NROWS / 256, 256, 0, stream>>>(E, posIdx, negIdx, rowLoss);
    reduce_kernel<<<1, 256, 0, stream>>>(rowLoss, (float*)d_out);
}